// ExponentialMovingAverage_57681410786025
// MI455X (gfx1250) — compile-verified
//
#include <hip/hip_runtime.h>
#include <cstdint>

// EMA scan over frames:  y[n] = w*x[n] + (1-w)*y[n-1]
// shapes: input (16,8,256,2048) f32, initial_state (16,8,256), weight (8,256)
#define N_BATCH   16
#define N_RES     8
#define N_BINS    256
#define N_FRAMES  2048
#define N_CHANNELS (N_BATCH * N_RES * N_BINS)   // 32768, one wave each
#define TILE      128                            // frames per wave iteration (4/lane)
#define N_TILES   (N_FRAMES / TILE)              // 16
#define WAVES_PER_BLOCK 8
#define THREADS   (WAVES_PER_BLOCK * 32)

__global__ __launch_bounds__(THREADS)
void ema_scan_async_kernel(const float* __restrict__ x,
                           const float* __restrict__ init_state,
                           const float* __restrict__ weight,
                           float* __restrict__ y)
{
    // per-wave double buffer: 2 * 128 floats = 1 KB / wave, 8 KB / block
    __shared__ __align__(16) float lds[WAVES_PER_BLOCK * 2 * TILE];

    const int lane = threadIdx.x & 31;
    const int wv   = threadIdx.x >> 5;
    const int c    = blockIdx.x * WAVES_PER_BLOCK + wv;   // channel id, grid sized exactly

    // weight is indexed by (res, bin) = c % (N_RES*N_BINS); clamp to [0,1]
    const float w_raw = weight[c & (N_RES * N_BINS - 1)];
    const float w  = fminf(fmaxf(w_raw, 0.0f), 1.0f);
    const float om = 1.0f - w;

    // wave-uniform scan coefficients: om^(4*2^s)
    const float om2 = om * om;
    const float om4 = om2 * om2;      // each lane composes 4 frames
    const float q1  = om4;            // shuffle distance 1  -> om^4
    const float q2  = q1 * q1;        // distance 2  -> om^8
    const float q4  = q2 * q2;        // distance 4  -> om^16
    const float q8  = q4 * q4;        // distance 8  -> om^32
    const float q16 = q8 * q8;        // distance 16 -> om^64

    // per-lane carry attenuation p = om^(4*lane), exact square-and-multiply
    float p = 1.0f;
    {
        float m = om4;
        if (lane & 1)  p *= m;  m *= m;
        if (lane & 2)  p *= m;  m *= m;
        if (lane & 4)  p *= m;  m *= m;
        if (lane & 8)  p *= m;  m *= m;
        if (lane & 16) p *= m;
    }

    float carry = init_state[c];

    const float* gbase = x + (size_t)c * N_FRAMES;
    float4*      ybase = (float4*)(y + (size_t)c * N_FRAMES);

    float* lwave = &lds[wv * 2 * TILE];
    // LDS byte offset of this lane's 16B slot in buffer 0 (flat addr low 32 bits == LDS offset)
    const unsigned lds_lane_byte =
        (unsigned)(uintptr_t)lwave + (unsigned)lane * 16u;

    // ---- prologue: async-fetch tile 0 into buffer 0 (ASYNCcnt path) ----
    {
        const float* gp = gbase + lane * 4;
        asm volatile("global_load_async_to_lds_b128 %0, %1, off"
                     :: "v"(lds_lane_byte), "v"(gp) : "memory");
    }

    for (int t = 0; t < N_TILES; ++t) {
        // issue next tile, then wait so exactly the in-flight prefetch remains
        if (t + 1 < N_TILES) {
            const float*   gp  = gbase + (t + 1) * TILE + lane * 4;
            const unsigned dst = lds_lane_byte + ((unsigned)(t + 1) & 1u) * (TILE * 4u);
            asm volatile("global_load_async_to_lds_b128 %0, %1, off"
                         :: "v"(dst), "v"(gp) : "memory");
            asm volatile("s_wait_asynccnt 0x1" ::: "memory");
        } else {
            asm volatile("s_wait_asynccnt 0x0" ::: "memory");
        }

        // consume current tile from LDS (ds_load_b128, conflict-free 16B/lane)
        const float4 xv = *(const float4*)(&lwave[(t & 1) * TILE + lane * 4]);

        // lane-local zero-carry response over 4 frames: b_j = om*b_{j-1} + w*x_j
        float b;
        b = w * xv.x;
        b = fmaf(om, b, w * xv.y);
        b = fmaf(om, b, w * xv.z);
        b = fmaf(om, b, w * xv.w);

        // Hillis-Steele inclusive scan of b across 32 lanes (wave-uniform multipliers)
        float s = b, up;
        up = __shfl_up(s, 1);  if (lane >= 1)  s = fmaf(q1,  up, s);
        up = __shfl_up(s, 2);  if (lane >= 2)  s = fmaf(q2,  up, s);
        up = __shfl_up(s, 4);  if (lane >= 4)  s = fmaf(q4,  up, s);
        up = __shfl_up(s, 8);  if (lane >= 8)  s = fmaf(q8,  up, s);
        up = __shfl_up(s, 16); if (lane >= 16) s = fmaf(q16, up, s);

        // exclusive prefix = state entering this lane with zero global carry
        float excl = __shfl_up(s, 1);
        if (lane == 0) excl = 0.0f;

        // y-state just before this lane's first frame: om^(4*lane)*carry + excl
        const float yprev = fmaf(p, carry, excl);

        // reconstruct exact per-frame outputs (matches reference recurrence form)
        float4 yo;
        yo.x = fmaf(om, yprev, w * xv.x);
        yo.y = fmaf(om, yo.x,  w * xv.y);
        yo.z = fmaf(om, yo.y,  w * xv.z);
        yo.w = fmaf(om, yo.z,  w * xv.w);

        ybase[t * 32 + lane] = yo;              // 512B coalesced store per wave

        carry = __shfl(yo.w, 31);               // carry into next 128-frame tile
    }
}

extern "C" void kernel_launch(void* const* d_in, const int* in_sizes, int n_in,
                              void* d_out, int out_size, void* d_ws, size_t ws_size,
                              hipStream_t stream) {
    (void)in_sizes; (void)n_in; (void)out_size; (void)d_ws; (void)ws_size;
    const float* x    = (const float*)d_in[0];
    const float* init = (const float*)d_in[1];
    const float* wt   = (const float*)d_in[2];
    float*       y    = (float*)d_out;

    const int blocks = N_CHANNELS / WAVES_PER_BLOCK;   // 4096
    ema_scan_async_kernel<<<blocks, THREADS, 0, stream>>>(x, init, wt, y);
}